// GAT_R_to_E_73839077752939
// MI455X (gfx1250) — compile-verified
//
#include <hip/hip_runtime.h>
#include <math.h>

// GAT relation->entity aggregation for MI455X (gfx1250, wave32).
// Stage 1 (node scores) uses V_WMMA_F32_16X16X4_F32: 16 node rows per wave,
// B matrix columns 0/1 carry a_h_w / a_t_w slices (branch-free via a
// 3-region LDS table: [a_h | a_t | zeros]). Everything else is the
// memory/atomic-bound part of the roofline (segment softmax + scatter-add).

typedef __attribute__((ext_vector_type(2))) float v2f;
typedef __attribute__((ext_vector_type(8))) float v8f;

#define EH 256   // entity hidden (K of the GEMV)
#define RH 64    // relation hidden (message width)

__device__ __forceinline__ unsigned enc_f32(float x) {
  // monotone map float -> unsigned so atomicMax(u32) orders like float
  unsigned u = __float_as_uint(x);
  return (u & 0x80000000u) ? ~u : (u | 0x80000000u);
}
__device__ __forceinline__ float dec_f32(unsigned k) {
  unsigned u = (k & 0x80000000u) ? (k ^ 0x80000000u) : ~k;
  return __uint_as_float(u);
}
__device__ __forceinline__ float leaky(float x) {
  return x > 0.0f ? x : 0.01f * x;
}

__global__ void zero_f32_kernel(float* p, long n) {
  long i = (long)blockIdx.x * blockDim.x + threadIdx.x;
  if (i < n) p[i] = 0.0f;
}
__global__ void fill_u32_kernel(unsigned* p, unsigned v, long n) {
  long i = (long)blockIdx.x * blockDim.x + threadIdx.x;
  if (i < n) p[i] = v;
}

// ---------------------------------------------------------------------------
// Stage 1: s_h[n] = x_e[n,:]·a_h,  s_t[n] = x_e[n,:]·a_t  via WMMA f32 16x16x4.
// One wave owns a 16-row tile. A tile: lane m (0..15) holds K=k+0,k+1 of row
// row0+m; lane m+16 holds K=k+2,k+3 (ISA 7.12.2, 32-bit A 16x4 layout).
// B tile (4x16): lanes 0..15 rows K=k+0 (b.x), k+1 (b.y); lanes 16..31 rows
// K=k+2, k+3. Column N=0 -> a_h, N=1 -> a_t, rest zero -- selected once per
// lane via a base pointer into the LDS table [a_h | a_t | zeros], so the
// inner loop is global_load_b64 + ds_load_b64 + v_wmma (no divergence).
// C: column N lives on lanes N (M=0..7 in c[0..7]) and N+16 (M=8..15).
// ---------------------------------------------------------------------------
__global__ void node_scores_wmma(const float* __restrict__ x_e,
                                 const float* __restrict__ a_h,
                                 const float* __restrict__ a_t,
                                 float* __restrict__ s_h,
                                 float* __restrict__ s_t,
                                 int n_tiles) {
  __shared__ float sB[3 * EH];            // [a_h | a_t | zeros]
  int tid = threadIdx.x;
  if (tid < EH) {
    sB[tid]          = a_h[tid];
    sB[EH + tid]     = a_t[tid];
    sB[2 * EH + tid] = 0.0f;
  }
  __syncthreads();

  int wave = tid >> 5;
  int lane = tid & 31;
  int tile = blockIdx.x * (blockDim.x >> 5) + wave;
  if (tile >= n_tiles) return;           // wave-uniform: EXEC all-1s below

  int row0  = tile * 16;
  int m     = lane & 15;                 // A: row within tile; B: column N
  int khalf = (lane < 16) ? 0 : 2;       // which K pair this half-wave holds
  const float* rowp = x_e + (size_t)(row0 + m) * EH + khalf;

  int sel = (m < 2) ? m : 2;             // column 0 -> a_h, 1 -> a_t, else zeros
  const float* bp = sB + sel * EH + khalf;

  v8f c = {};
#pragma unroll 4
  for (int k = 0; k < EH; k += 4) {
    float2 av = *(const float2*)(rowp + k);  // 8B aligned (khalf in {0,2}, k%4==0)
    float2 bv = *(const float2*)(bp + k);    // single ds_load_b64, branch-free
    v2f a; a.x = av.x; a.y = av.y;
    v2f b; b.x = bv.x; b.y = bv.y;
    c = __builtin_amdgcn_wmma_f32_16x16x4_f32(false, a, false, b,
                                              (short)0, c, false, false);
  }

  if (lane == 0) {        // column N=0, rows M=0..7  -> s_h
    s_h[row0+0]=c[0]; s_h[row0+1]=c[1]; s_h[row0+2]=c[2]; s_h[row0+3]=c[3];
    s_h[row0+4]=c[4]; s_h[row0+5]=c[5]; s_h[row0+6]=c[6]; s_h[row0+7]=c[7];
  } else if (lane == 16) { // column N=0, rows M=8..15
    s_h[row0+8]=c[0];  s_h[row0+9]=c[1];  s_h[row0+10]=c[2]; s_h[row0+11]=c[3];
    s_h[row0+12]=c[4]; s_h[row0+13]=c[5]; s_h[row0+14]=c[6]; s_h[row0+15]=c[7];
  } else if (lane == 1) {  // column N=1 -> s_t
    s_t[row0+0]=c[0]; s_t[row0+1]=c[1]; s_t[row0+2]=c[2]; s_t[row0+3]=c[3];
    s_t[row0+4]=c[4]; s_t[row0+5]=c[5]; s_t[row0+6]=c[6]; s_t[row0+7]=c[7];
  } else if (lane == 17) {
    s_t[row0+8]=c[0];  s_t[row0+9]=c[1];  s_t[row0+10]=c[2]; s_t[row0+11]=c[3];
    s_t[row0+12]=c[4]; s_t[row0+13]=c[5]; s_t[row0+14]=c[6]; s_t[row0+15]=c[7];
  }
}

// scalar fallback for a partial 16-row tail tile (empty for N_e=50000)
__global__ void node_scores_tail(const float* __restrict__ x_e,
                                 const float* __restrict__ a_h,
                                 const float* __restrict__ a_t,
                                 float* s_h, float* s_t,
                                 int row_start, int n_e) {
  int r = row_start + blockIdx.x * blockDim.x + threadIdx.x;
  if (r >= n_e) return;
  float ah = 0.0f, at = 0.0f;
  const float* row = x_e + (size_t)r * EH;
  for (int k = 0; k < EH; ++k) { float x = row[k]; ah += x*a_h[k]; at += x*a_t[k]; }
  s_h[r] = ah; s_t[r] = at;
}

// Stage 2: s_r[r] = x_r[r,:]·a_r  (wave per relation, 2 elems per lane)
__global__ void rel_scores(const float* __restrict__ x_r,
                           const float* __restrict__ a_r,
                           float* __restrict__ s_r, int n_r) {
  int wave = threadIdx.x >> 5, lane = threadIdx.x & 31;
  int r = blockIdx.x * (blockDim.x >> 5) + wave;
  if (r >= n_r) return;
  const float* row = x_r + (size_t)r * RH;
  float v = row[lane] * a_r[lane] + row[32 + lane] * a_r[32 + lane];
  for (int off = 16; off; off >>= 1) v += __shfl_xor(v, off, 32);
  if (lane == 0) s_r[r] = v;
}

// Stage 3: per-edge logits + segment max via monotone-u32 atomicMax
__global__ void edge_logits_max(const int* __restrict__ head,
                                const int* __restrict__ tail,
                                const int* __restrict__ rel,
                                const float* __restrict__ s_h,
                                const float* __restrict__ s_t,
                                const float* __restrict__ s_r,
                                float* __restrict__ lh, float* __restrict__ lt,
                                unsigned* __restrict__ maxh,
                                unsigned* __restrict__ maxt, int ne) {
  int e = blockIdx.x * blockDim.x + threadIdx.x;
  if (e >= ne) return;
  int h = head[e], t = tail[e], r = rel[e];
  float er = s_r[r];
  float vh = leaky(s_h[h] + er);
  float vt = leaky(s_t[t] + er);
  lh[e] = vh; lt[e] = vt;
  atomicMax(&maxh[h], enc_f32(vh));
  atomicMax(&maxt[t], enc_f32(vt));
}

// Stage 4: ex = exp(logit - segmax); accumulate per-segment denominators
__global__ void edge_exp_den(const int* __restrict__ head,
                             const int* __restrict__ tail,
                             float* __restrict__ lh, float* __restrict__ lt,
                             const unsigned* __restrict__ maxh,
                             const unsigned* __restrict__ maxt,
                             float* __restrict__ denh, float* __restrict__ dent,
                             int ne) {
  int e = blockIdx.x * blockDim.x + threadIdx.x;
  if (e >= ne) return;
  int h = head[e], t = tail[e];
  float exh = __expf(lh[e] - dec_f32(maxh[h]));
  float ext = __expf(lt[e] - dec_f32(maxt[t]));
  lh[e] = exh; lt[e] = ext;               // reuse logit buffers for exp
  atomicAdd(&denh[h], exh);
  atomicAdd(&dent[t], ext);
}

// Stage 5: wave per edge; lane d handles dims {d, d+32}. 4 f32 atomics/lane.
__global__ void scatter_out(const int* __restrict__ head,
                            const int* __restrict__ tail,
                            const int* __restrict__ rel,
                            const float* __restrict__ exh,
                            const float* __restrict__ ext,
                            const float* __restrict__ denh,
                            const float* __restrict__ dent,
                            const float* __restrict__ x_r,
                            float* __restrict__ out, int ne) {
  int wave = threadIdx.x >> 5, lane = threadIdx.x & 31;
  long e = (long)blockIdx.x * (blockDim.x >> 5) + wave;
  if (e >= ne) return;
  int h = head[e], t = tail[e], r = rel[e];
  float ah = exh[e] / (denh[h] + 1e-16f);
  float at = ext[e] / (dent[t] + 1e-16f);
  float m0 = x_r[(size_t)r * RH + lane];
  float m1 = x_r[(size_t)r * RH + 32 + lane];
  float* oh = out + (size_t)h * (2 * RH);       // head half: cols [0,64)
  float* ot = out + (size_t)t * (2 * RH) + RH;  // tail half: cols [64,128)
  atomicAdd(&oh[lane],      ah * m0);
  atomicAdd(&oh[32 + lane], ah * m1);
  atomicAdd(&ot[lane],      at * m0);
  atomicAdd(&ot[32 + lane], at * m1);
}

extern "C" void kernel_launch(void* const* d_in, const int* in_sizes, int n_in,
                              void* d_out, int out_size, void* d_ws, size_t ws_size,
                              hipStream_t stream) {
  const float* x_e = (const float*)d_in[0];
  const float* x_r = (const float*)d_in[1];
  const int*   edge_index = (const int*)d_in[2];
  const int*   rel = (const int*)d_in[3];
  // d_in[4] line_graph_index, d_in[5] line_graph_val: unused by reference
  const float* a_h = (const float*)d_in[6];
  const float* a_t = (const float*)d_in[7];
  const float* a_r = (const float*)d_in[8];

  const int n_e = in_sizes[0] / EH;
  const int n_r = in_sizes[1] / RH;
  const int E   = in_sizes[3];
  const int* head = edge_index;
  const int* tail = edge_index + E;

  // workspace carve-up
  float*    s_h  = (float*)d_ws;
  float*    s_t  = s_h + n_e;
  float*    s_r  = s_t + n_e;
  unsigned* maxh = (unsigned*)(s_r + n_r);
  unsigned* maxt = maxh + n_e;
  float*    denh = (float*)(maxt + n_e);
  float*    dent = denh + n_e;
  float*    lh   = dent + n_e;   // logits -> exp(logit - max)
  float*    lt   = lh + E;

  // init: zero output + denominators, "-inf" (encoded 0) segment maxima
  long outN = (long)n_e * 2 * RH;
  zero_f32_kernel<<<(int)((outN + 255) / 256), 256, 0, stream>>>((float*)d_out, outN);
  zero_f32_kernel<<<(int)((2L*n_e + 255) / 256), 256, 0, stream>>>(denh, 2L*n_e);
  fill_u32_kernel<<<(int)((2L*n_e + 255) / 256), 256, 0, stream>>>(maxh, 0u, 2L*n_e);

  // stage 1: node scores (WMMA tiles + scalar tail)
  int n_tiles = n_e / 16;
  if (n_tiles > 0)
    node_scores_wmma<<<(n_tiles + 7) / 8, 256, 0, stream>>>(x_e, a_h, a_t, s_h, s_t, n_tiles);
  int row_start = n_tiles * 16;
  if (row_start < n_e)
    node_scores_tail<<<(n_e - row_start + 255) / 256, 256, 0, stream>>>(
        x_e, a_h, a_t, s_h, s_t, row_start, n_e);

  // stage 2: relation scores
  rel_scores<<<(n_r + 7) / 8, 256, 0, stream>>>(x_r, a_r, s_r, n_r);

  // stage 3..5: segment softmax + scatter
  edge_logits_max<<<(E + 255) / 256, 256, 0, stream>>>(head, tail, rel, s_h, s_t, s_r,
                                                       lh, lt, maxh, maxt, E);
  edge_exp_den<<<(E + 255) / 256, 256, 0, stream>>>(head, tail, lh, lt, maxh, maxt,
                                                    denh, dent, E);
  scatter_out<<<(E + 7) / 8, 256, 0, stream>>>(head, tail, rel, lh, lt, denh, dent,
                                               x_r, (float*)d_out, E);
}